// KernalAnsatz_65481071398549
// MI455X (gfx1250) — compile-verified
//
#include <hip/hip_runtime.h>
#include <hip/hip_bf16.h>

typedef __attribute__((ext_vector_type(2))) float v2f;
typedef __attribute__((ext_vector_type(8))) float v8f;

#define NROWS 8192
#define KDIM  64
#define KSTEPS 16   // KDIM / 4

// ---------------------------------------------------------------------------
// Row-norm precompute: out[r] = sum_k M[r][k]^2  (one thread per row)
// ---------------------------------------------------------------------------
__global__ __launch_bounds__(256) void rbf_row_norms(const float* __restrict__ M,
                                                     float* __restrict__ out,
                                                     int nrows) {
    int r = blockIdx.x * blockDim.x + threadIdx.x;
    if (r >= nrows) return;
    const float4* p = (const float4*)(M + (size_t)r * KDIM);
    float s = 0.0f;
#pragma unroll
    for (int i = 0; i < KDIM / 4; ++i) {
        float4 v = p[i];
        s += v.x * v.x + v.y * v.y + v.z * v.z + v.w * v.w;
    }
    out[r] = s;
}

// ---------------------------------------------------------------------------
// Main kernel: each wave computes one 16-row M-tile x four 16-col N-tiles of
// Out = exp(-(x2 + y2 - 2*X.Y^T)) using V_WMMA_F32_16X16X4_F32.
//
// f32 WMMA fragment layout (CDNA5 ISA, 32-bit A 16x4):
//   lanes 0-15 : VGPR0=K(k0), VGPR1=K(k0+1), row M = lane
//   lanes 16-31: VGPR0=K(k0+2), VGPR1=K(k0+3), row M = lane-16
// B (4x16) is symmetric over Y rows since B[k][n] = Y[n][k].
// C/D 16x16 f32: VGPR r holds M=r (lanes 0-15) / M=r+8 (lanes 16-31), N=lane&15.
// ---------------------------------------------------------------------------
__global__ __launch_bounds__(256) void rbf_wmma_f32(const float* __restrict__ X,
                                                    const float* __restrict__ Y,
                                                    const float* __restrict__ x2,
                                                    const float* __restrict__ y2,
                                                    float* __restrict__ Out) {
    const int lane    = threadIdx.x & 31;
    const int wave    = threadIdx.x >> 5;          // 0..7
    const int tm      = blockIdx.y * 8 + wave;     // M-tile 0..511
    const int tn0     = blockIdx.x * 4;            // first of 4 N-tiles
    const int halfSel = lane >> 4;                 // 0: lanes 0-15, 1: lanes 16-31
    const int r16     = lane & 15;
    const int kOff    = halfSel * 2;               // K sub-offset per half-wave

    // ---- A panel for this wave's 16 rows of X: 16 k-steps, reused 4x ----
    const int   rowA = tm * 16 + r16;
    const float* xrp = X + (size_t)rowA * KDIM + kOff;
    v2f a[KSTEPS];
#pragma unroll
    for (int s = 0; s < KSTEPS; ++s)
        a[s] = *(const v2f*)(xrp + 4 * s);

    const float x2l = x2[rowA];                    // x2 for row (tm*16 + r16)

    for (int t = 0; t < 4; ++t) {
        const int    tn   = tn0 + t;
        const int    colB = tn * 16 + r16;
        const float* yrp  = Y + (size_t)colB * KDIM + kOff;

        if (t < 3) {  // prefetch next Y panel (gfx1250 global_prefetch_b8)
            __builtin_prefetch(yrp + KDIM * 16, 0, 1);
        }

        v8f c = {};
#pragma unroll
        for (int s = 0; s < KSTEPS; ++s) {
            v2f b = *(const v2f*)(yrp + 4 * s);
            // emits v_wmma_f32_16x16x4_f32
            c = __builtin_amdgcn_wmma_f32_16x16x4_f32(
                    /*neg_a=*/false, a[s], /*neg_b=*/false, b,
                    /*c_mod=*/(short)0, c, /*reuse_a=*/false, /*reuse_b=*/false);
        }

        // ---- fused epilogue: dist2 -> clamp -> exp -> NT store ----
        const float y2v = y2[colB];
        float* outp = Out + (size_t)(tm * 16 + halfSel * 8) * NROWS + colB;
#pragma unroll
        for (int r = 0; r < 8; ++r) {
            // x2 of the output row this accumulator register maps to
            float xx = __shfl(x2l, r + halfSel * 8, 32);
            float d  = xx + y2v - 2.0f * c[r];
            d        = fmaxf(d, 0.0f);
            float v  = __expf(-d);                 // GAMMA = 1.0
            __builtin_nontemporal_store(v, outp + (size_t)r * NROWS);
        }
    }
}

// ---------------------------------------------------------------------------
extern "C" void kernel_launch(void* const* d_in, const int* in_sizes, int n_in,
                              void* d_out, int out_size, void* d_ws, size_t ws_size,
                              hipStream_t stream) {
    const float* X = (const float*)d_in[0];
    const float* Y = (const float*)d_in[1];
    float* x2 = (float*)d_ws;            // 8192 floats
    float* y2 = x2 + NROWS;              // 8192 floats  (64 KB total in d_ws)
    float* O  = (float*)d_out;

    rbf_row_norms<<<NROWS / 256, 256, 0, stream>>>(X, x2, NROWS);
    rbf_row_norms<<<NROWS / 256, 256, 0, stream>>>(Y, y2, NROWS);

    // 512x512 tiles of 16x16; block = 8 waves (128 rows) x 4 N-tiles (64 cols)
    dim3 grid(NROWS / 64, NROWS / 128);  // (128, 64)
    rbf_wmma_f32<<<grid, 256, 0, stream>>>(X, Y, x2, y2, O);
}